// DigitCapsules_67053029425776
// MI455X (gfx1250) — compile-verified
//
#include <hip/hip_runtime.h>
#include <hip/hip_bf16.h>

// CapsNet dynamic routing, fully fused per (out_capsule o, 16-batch tile).
// priors are never materialized in HBM (saves ~1.9 GB of traffic); they are
// recomputed from L2-resident x / route_weights via chained
// V_WMMA_F32_16X16X4_F32 in every routing pass.

typedef __attribute__((ext_vector_type(2))) float v2f;
typedef __attribute__((ext_vector_type(8))) float v8f;

#define O_CAPS   10
#define I_CAPS   1152
#define D_IN     8
#define E_OUT    16
#define BATCH    512
#define BT       16          // batch tile (WMMA M)
#define NWAVES   8
#define NTHREADS 256         // 8 x wave32
#define NITER    3

// dynamic LDS partition (floats)
#define LDS_LOGITS  0                       // [I_CAPS][BT]       18432 f
#define LDS_PART    (I_CAPS * BT)           // [NWAVES][16][16]    2048 f
#define LDS_RED     (LDS_PART + NWAVES*256) // [16][16]             256 f
#define LDS_OUT     (LDS_RED + 256)         // [16][16]             256 f
#define LDS_MAXB    (LDS_OUT + 256)         // [16]
#define LDS_INVD    (LDS_MAXB + 16)         // [16]
#define LDS_SCALE   (LDS_INVD + 16)         // [16]
#define LDS_FLOATS  (LDS_SCALE + 16)

__global__ __launch_bounds__(NTHREADS)
void caps_route_kernel(const float* __restrict__ x,    // [B][I][D]
                       const float* __restrict__ w,    // [O][I][D][E]
                       float* __restrict__ out)        // [O][B][E]
{
    extern __shared__ float lds[];

    const int t    = threadIdx.x;
    const int lane = t & 31;
    const int wv   = t >> 5;
    const int half = lane >> 4;   // which K-pair this lane supplies to WMMA A/B
    const int l16  = lane & 15;

    const int o  = blockIdx.x / (BATCH / BT);
    const int bt = blockIdx.x % (BATCH / BT);
    const int b0 = bt * BT;

    // logits start at zero
    for (int k = t; k < I_CAPS * BT; k += NTHREADS) lds[LDS_LOGITS + k] = 0.0f;
    __syncthreads();

    const float* xb = x + (size_t)(b0 + l16) * I_CAPS * D_IN;  // this lane's batch row
    const float* wo = w + (size_t)o * I_CAPS * D_IN * E_OUT;

    for (int it = 0; it < NITER; ++it) {
        // ---------- per-b softmax stats over i (max, 1/denom) ----------
        {
            const int b = t & 15, g = t >> 4;
            float m = -3.0e38f;
            for (int i = g; i < I_CAPS; i += 16)
                m = fmaxf(m, lds[LDS_LOGITS + i * BT + b]);
            lds[LDS_PART + g * 16 + b] = m;
            __syncthreads();
            if (t < 16) {
                float mm = lds[LDS_PART + t];
                for (int g2 = 1; g2 < 16; ++g2)
                    mm = fmaxf(mm, lds[LDS_PART + g2 * 16 + t]);
                lds[LDS_MAXB + t] = mm;
            }
            __syncthreads();
            const float mb = lds[LDS_MAXB + b];
            float s = 0.0f;
            for (int i = g; i < I_CAPS; i += 16)
                s += __expf(lds[LDS_LOGITS + i * BT + b] - mb);
            lds[LDS_PART + g * 16 + b] = s;
            __syncthreads();
            if (t < 16) {
                float ss = 0.0f;
                for (int g2 = 0; g2 < 16; ++g2) ss += lds[LDS_PART + g2 * 16 + t];
                lds[LDS_INVD + t] = 1.0f / ss;
            }
            __syncthreads();
        }

        // ---------- s-pass: s[b,e] = sum_{i,d} (p_ib * x[b,i,d]) * w[o,i,d,e]
        // chained WMMA 16x16x4 f32, M=b, N=e, K streams over (i,d) ----------
        {
            v8f acc = {};
            const float mb  = lds[LDS_MAXB + l16];
            const float ivd = lds[LDS_INVD + l16];
            const int i_beg = wv * (I_CAPS / NWAVES);
            const int i_end = i_beg + (I_CAPS / NWAVES);
            for (int i = i_beg; i < i_end; ++i) {
                const float p  = __expf(lds[LDS_LOGITS + i * BT + l16] - mb) * ivd;
                const float* xr = xb + (size_t)i * D_IN;           // 8 contiguous f32
                const float* wr = wo + (size_t)i * D_IN * E_OUT;   // [d][e]
                // A (16x4 f32): lane supplies K = 2*half+{0,1}
                v2f a1, a2, bb1, bb2;
                a1.x = p * xr[2 * half + 0];
                a1.y = p * xr[2 * half + 1];
                a2.x = p * xr[4 + 2 * half + 0];
                a2.y = p * xr[4 + 2 * half + 1];
                // B (4x16 f32): vgpr0 = rows {0,2}, vgpr1 = rows {1,3}
                bb1.x = wr[(2 * half + 0) * E_OUT + l16];
                bb1.y = wr[(2 * half + 1) * E_OUT + l16];
                bb2.x = wr[(4 + 2 * half + 0) * E_OUT + l16];
                bb2.y = wr[(4 + 2 * half + 1) * E_OUT + l16];
                acc = __builtin_amdgcn_wmma_f32_16x16x4_f32(false, a1, false, bb1,
                                                            (short)0, acc, false, false);
                acc = __builtin_amdgcn_wmma_f32_16x16x4_f32(false, a2, false, bb2,
                                                            (short)0, acc, false, false);
            }
            // C layout: vgpr r -> row M = r + 8*half, col N = l16
            for (int r = 0; r < 8; ++r)
                lds[LDS_PART + wv * 256 + (r + 8 * half) * 16 + l16] = acc[r];
        }
        __syncthreads();

        // ---------- cross-wave reduce + squash ----------
        {
            const int b = t >> 4, e = t & 15;
            float s = 0.0f;
            for (int w2 = 0; w2 < NWAVES; ++w2)
                s += lds[LDS_PART + w2 * 256 + b * 16 + e];
            lds[LDS_RED + b * 16 + e] = s;
        }
        __syncthreads();
        if (t < 16) {
            float sq = 0.0f;
            for (int e = 0; e < 16; ++e) {
                const float v = lds[LDS_RED + t * 16 + e];
                sq += v * v;
            }
            lds[LDS_SCALE + t] = sq / ((1.0f + sq) * sqrtf(sq));
        }
        __syncthreads();
        {
            const int b = t >> 4, e = t & 15;
            const float v = lds[LDS_RED + b * 16 + e] * lds[LDS_SCALE + b];
            lds[LDS_OUT + b * 16 + e] = v;
            if (it == NITER - 1)
                out[((size_t)o * BATCH + b0 + b) * E_OUT + e] = v;  // [O,B,1,1,E]
        }
        __syncthreads();

        if (it == NITER - 1) break;

        // ---------- delta-pass: T[(i_pair,d), b] = W[i,d,:] . out^T  (WMMA),
        // then in-lane delta[i,b] = sum_d x[b,i,d] * T, logits += delta -------
        {
            // B = out^T (shared across i): chunk c covers e-rows 4c..4c+3
            v2f bts[4];
            for (int c = 0; c < 4; ++c) {
                bts[c].x = lds[LDS_OUT + l16 * 16 + (4 * c + 2 * half + 0)];
                bts[c].y = lds[LDS_OUT + l16 * 16 + (4 * c + 2 * half + 1)];
            }
            const int m     = l16;
            const int dA    = m & 7;     // A-row's d
            const int iselA = m >> 3;    // rows 0-7 -> i0, 8-15 -> i1
            const int jb    = wv * (I_CAPS / 2 / NWAVES);
            const int je    = jb + (I_CAPS / 2 / NWAVES);
            for (int j = jb; j < je; ++j) {
                const int i0 = 2 * j;
                v8f c8 = {};
                const float* wrA =
                    wo + ((size_t)(i0 + iselA) * D_IN + dA) * E_OUT;
                for (int c = 0; c < 4; ++c) {
                    v2f a;
                    a.x = wrA[4 * c + 2 * half + 0];
                    a.y = wrA[4 * c + 2 * half + 1];
                    c8 = __builtin_amdgcn_wmma_f32_16x16x4_f32(false, a, false, bts[c],
                                                               (short)0, c8, false, false);
                }
                // lane holds T[M = r+8*half, b=l16]; M<8 -> i0 (d=M), M>=8 -> i1
                const int isel  = i0 + half;
                const float* xr = xb + (size_t)isel * D_IN;
                float delta = 0.0f;
                for (int r = 0; r < 8; ++r) delta += c8[r] * xr[r];
                lds[LDS_LOGITS + isel * BT + l16] += delta;
            }
        }
        __syncthreads();
    }
}

extern "C" void kernel_launch(void* const* d_in, const int* in_sizes, int n_in,
                              void* d_out, int out_size, void* d_ws, size_t ws_size,
                              hipStream_t stream) {
    (void)in_sizes; (void)n_in; (void)d_ws; (void)ws_size; (void)out_size;
    const float* x = (const float*)d_in[0];           // [512,1152,8]
    const float* w = (const float*)d_in[1];           // [10,1152,8,16]
    float* out     = (float*)d_out;                   // [10,512,1,1,16]

    const int lds_bytes = LDS_FLOATS * (int)sizeof(float);   // ~84 KB (<320 KB/WGP)
    (void)hipFuncSetAttribute((const void*)caps_route_kernel,
                              hipFuncAttributeMaxDynamicSharedMemorySize, lds_bytes);

    dim3 grid(O_CAPS * (BATCH / BT));   // 320 workgroups
    dim3 block(NTHREADS);               // 8 x wave32
    caps_route_kernel<<<grid, block, lds_bytes, stream>>>(x, w, out);
}